// InfNet_13365938225801
// MI455X (gfx1250) — compile-verified
//
#include <hip/hip_runtime.h>
#include <hip/hip_bf16.h>

typedef float v2f __attribute__((ext_vector_type(2)));
typedef float v8f __attribute__((ext_vector_type(8)));

#define N_NODES   100000
#define N_EDGES   1600000
#define IN_DIM    128
#define NUM_MASK  80000
#define PERM_A    73939133u   // odd, not /5 -> coprime with 100000 => exact permutation
#define STRIDE64  64

// ---------------- mask / degree / norm prep ----------------

__global__ void init_flag_deg(int* __restrict__ flag, float* __restrict__ deg, int n) {
    int i = blockIdx.x * blockDim.x + threadIdx.x;
    if (i < n) { flag[i] = 1; deg[i] = 1.0f; }  // deg starts at 1 (self loop)
}

__global__ void build_mask(int* __restrict__ flag, float* __restrict__ mask_out, int nmask, int n) {
    int i = blockIdx.x * blockDim.x + threadIdx.x;
    if (i < nmask) {
        unsigned node = (unsigned)(((unsigned long long)i * PERM_A) % (unsigned long long)n);
        flag[node] = 0;
        mask_out[i] = (float)node;
    }
}

__global__ void accum_deg(const int* __restrict__ dst, float* __restrict__ deg, int ne) {
    int e = blockIdx.x * blockDim.x + threadIdx.x;
    if (e < ne) atomicAdd(&deg[dst[e]], 1.0f);
}

__global__ void deg_to_dinv(float* __restrict__ deg, int n) {
    int i = blockIdx.x * blockDim.x + threadIdx.x;
    if (i < n) deg[i] = __frsqrt_rn(deg[i]);   // deg >= 1 always
}

__global__ void build_norm(const int* __restrict__ src, const int* __restrict__ dst,
                           const float* __restrict__ dinv, float* __restrict__ nrm, int ne) {
    int e = blockIdx.x * blockDim.x + threadIdx.x;
    if (e < ne) nrm[e] = dinv[src[e]] * dinv[dst[e]];
}

// ---------------- dense GEMM: T = A @ W  (WMMA f32 16x16x4) ----------------
// Also writes AGG = T * dinv^2 (the self-loop contribution seed).
// Block = 2 row-tiles x CT col-tiles of waves; each wave owns a 16x16 D tile.
// A: row-major [N x K]; W: row-major [K x Wcols], staged into LDS padded to CT*16.

template <int K, int CT, bool MASKED>
__global__ void gemm_wmma(const float* __restrict__ A, const float* __restrict__ Wg,
                          const int* __restrict__ flag, const float* __restrict__ token,
                          const float* __restrict__ dinv,
                          float* __restrict__ T, float* __restrict__ AGG, int wcols) {
    constexpr int CP = CT * 16;            // padded weight columns
    __shared__ float wlds[K * CP];

    const int tid = threadIdx.x;
    // stage W into LDS (zero-pad columns >= wcols)
    for (int idx = tid; idx < K * CP; idx += blockDim.x) {
        int kk = idx / CP, cc = idx % CP;
        wlds[idx] = (cc < wcols) ? Wg[kk * wcols + cc] : 0.0f;
    }
    __syncthreads();

    const int wid     = tid >> 5;
    const int lane    = tid & 31;
    const int waveRow = wid / CT;
    const int waveCol = wid % CT;
    const int r       = lane & 15;         // M (for A) / N (for B,D)
    const int kh      = lane >> 4;         // which K-pair this half-wave holds

    const int row0 = blockIdx.x * 32 + waveRow * 16;   // 100000 % 32 == 0: no guard
    const int col  = waveCol * 16 + r;

    const float* arow = A + (size_t)(row0 + r) * K;
    if (MASKED) {
        if (!flag[row0 + r]) arow = token;             // masked row -> enc_mask_token
    }

    v8f acc = {};
    #pragma unroll
    for (int kk = 0; kk < K; kk += 4) {
        const int kr = kk + kh * 2;
        v2f a, b;
        a.x = arow[kr];
        a.y = arow[kr + 1];
        b.x = wlds[kr * CP + col];
        b.y = wlds[(kr + 1) * CP + col];
        acc = __builtin_amdgcn_wmma_f32_16x16x4_f32(false, a, false, b,
                                                    (short)0, acc, false, false);
    }

    // D layout: vgpr rr -> M = rr (+8 for upper half-wave), N = lane&15
    const int rowbase = row0 + kh * 8;
    #pragma unroll
    for (int rr = 0; rr < 8; ++rr) {
        const int row = rowbase + rr;
        const size_t idx = (size_t)row * STRIDE64 + waveCol * 16 + r;
        const float v  = acc[rr];
        const float dv = dinv[row];
        T[idx]   = v;
        AGG[idx] = v * dv * dv;            // self-loop contribution
    }
}

// ---------------- edge scatter: AGG[dst] += T[src] * norm ----------------

template <int COLS>
__global__ void edge_scatter(const float* __restrict__ T,
                             const int* __restrict__ src, const int* __restrict__ dst,
                             const float* __restrict__ nrm, float* __restrict__ AGG, int ne) {
    constexpr int TPE = (COLS + 3) / 4;
    long long gid = (long long)blockIdx.x * blockDim.x + threadIdx.x;
    long long tot = (long long)ne * TPE;
    if (gid >= tot) return;
    const int e  = (int)(gid / TPE);
    const int c0 = (int)(gid % TPE) * 4;
    const int s = src[e], d = dst[e];
    const float w = nrm[e];
    const float* tp = T + (size_t)s * STRIDE64 + c0;
    float*       ap = AGG + (size_t)d * STRIDE64 + c0;
    if (c0 + 4 <= COLS) {
        const float4 v = *(const float4*)tp;   // 16B-aligned: row stride 64 floats
        atomicAdd(&ap[0], v.x * w);
        atomicAdd(&ap[1], v.y * w);
        atomicAdd(&ap[2], v.z * w);
        atomicAdd(&ap[3], v.w * w);
    } else {
        #pragma unroll
        for (int j = 0; j < 4; ++j)
            if (c0 + j < COLS) atomicAdd(&ap[j], tp[j] * w);
    }
}

// ---------------- softplus: ACT = log1p(exp(AGG)) ----------------

__global__ void softplus_k(const float* __restrict__ AGG, float* __restrict__ ACT,
                           int m, long long total) {
    long long t = (long long)blockIdx.x * blockDim.x + threadIdx.x;
    if (t >= total) return;
    const int i  = (int)(t / m);
    const int mm = (int)(t % m);
    const size_t idx = (size_t)i * STRIDE64 + mm;
    const float x = AGG[idx];
    ACT[idx] = fmaxf(x, 0.0f) + log1pf(expf(-fabsf(x)));
}

// ---------------- epilogue: z = lbd * Gamma(1 + 1/kappa) ----------------

__global__ void epilogue_k(const float* __restrict__ H, float* __restrict__ out, int n) {
    int i = blockIdx.x * blockDim.x + threadIdx.x;
    if (i >= n) return;
    const float* h = H + (size_t)i * STRIDE64;
    const float kap = h[32] + 0.1f;
    const float gam = expf(lgammaf(1.0f + 1.0f / kap));
    float* zout = out;                                  // [N,32]
    float* lout = out + (size_t)n * 32;                 // [N,32]
    float* kout = out + (size_t)n * 64;                 // [N,1]
    kout[i] = kap;
    #pragma unroll
    for (int m = 0; m < 32; ++m) {
        const float l = h[m];
        lout[(size_t)i * 32 + m] = l;
        zout[(size_t)i * 32 + m] = l * gam;
    }
}

// ---------------- launch ----------------

extern "C" void kernel_launch(void* const* d_in, const int* in_sizes, int n_in,
                              void* d_out, int out_size, void* d_ws, size_t ws_size,
                              hipStream_t stream) {
    (void)n_in; (void)out_size; (void)ws_size;
    const float* x     = (const float*)d_in[0];
    const int*   ei    = (const int*)d_in[1];
    const float* token = (const float*)d_in[2];
    const float* W0    = (const float*)d_in[3];
    const float* W1    = (const float*)d_in[4];
    const float* W2    = (const float*)d_in[5];
    float* out = (float*)d_out;

    const int N  = N_NODES;
    const int E  = in_sizes[1] / 2;
    const int* src = ei;
    const int* dst = ei + E;

    // workspace carve (floats)
    float* wsf = (float*)d_ws;
    size_t o = 0;
    int*   flag = (int*)(wsf + o); o += N;
    float* dinv = wsf + o;         o += N;        // holds deg, then rsqrt in place
    float* nrm  = wsf + o;         o += E;
    float* T    = wsf + o;         o += (size_t)N * STRIDE64;
    float* AGG  = wsf + o;         o += (size_t)N * STRIDE64;
    float* ACT  = wsf + o;         o += (size_t)N * STRIDE64;

    const int B = 256;
    auto blks = [](long long t, int b) { return (unsigned)((t + b - 1) / b); };

    // prep
    init_flag_deg<<<blks(N, B), B, 0, stream>>>(flag, dinv, N);
    build_mask<<<blks(NUM_MASK, B), B, 0, stream>>>(flag, out + (size_t)N * 65, NUM_MASK, N);
    accum_deg<<<blks(E, B), B, 0, stream>>>(dst, dinv, E);
    deg_to_dinv<<<blks(N, B), B, 0, stream>>>(dinv, N);
    build_norm<<<blks(E, B), B, 0, stream>>>(src, dst, dinv, nrm, E);

    const unsigned rowBlocks = N / 32;   // 3125, exact

    // layer 0: K=128 -> 64 cols, masked input
    gemm_wmma<128, 4, true><<<rowBlocks, 256, 0, stream>>>(x, W0, flag, token, dinv, T, AGG, 64);
    edge_scatter<64><<<blks((long long)E * 16, B), B, 0, stream>>>(T, src, dst, nrm, AGG, E);
    softplus_k<<<blks((long long)N * 64, B), B, 0, stream>>>(AGG, ACT, 64, (long long)N * 64);

    // layer 1: K=64 -> 64 cols
    gemm_wmma<64, 4, false><<<rowBlocks, 256, 0, stream>>>(ACT, W1, nullptr, nullptr, dinv, T, AGG, 64);
    edge_scatter<64><<<blks((long long)E * 16, B), B, 0, stream>>>(T, src, dst, nrm, AGG, E);
    softplus_k<<<blks((long long)N * 64, B), B, 0, stream>>>(AGG, ACT, 64, (long long)N * 64);

    // layer 2: K=64 -> 33 cols (padded to 48 in-tile)
    gemm_wmma<64, 3, false><<<rowBlocks, 192, 0, stream>>>(ACT, W2, nullptr, nullptr, dinv, T, AGG, 33);
    edge_scatter<33><<<blks((long long)E * 9, B), B, 0, stream>>>(T, src, dst, nrm, AGG, E);
    softplus_k<<<blks((long long)N * 33, B), B, 0, stream>>>(AGG, ACT, 33, (long long)N * 33);

    // outputs: z[N,32], lbd[N,32], kappa[N], mask_nodes[80000]
    epilogue_k<<<blks(N, B), B, 0, stream>>>(ACT, out, N);
}